// LaplacianLoss_11897059410700
// MI455X (gfx1250) — compile-verified
//
#include <hip/hip_runtime.h>

// Problem constants (fixed by the reference module)
#define NV 6890          // vertices (M and K of the GEMM)
#define NB 64            // batch
#define ND 3             // coord dim
#define NCOLS (NB * ND)  // 192 = GEMM N
#define KB 64            // K chunk staged in LDS per iteration
#define PAD 4            // LDS row padding (floats) to spread banks
#define ROWSTRIDE (KB + PAD)
#define MROWS 32         // rows of L per block (2 x 16-row M tiles per wave)
#define NCHUNK ((NV + KB - 1) / KB)   // 108
#define KTAIL ((NCHUNK - 1) * KB)     // 6848 (tail chunk has 42 valid cols)

typedef __attribute__((ext_vector_type(2))) float v2f;
typedef __attribute__((ext_vector_type(4))) float v4f;
typedef __attribute__((ext_vector_type(8))) float v8f;

__global__ void lap_zero_kernel(float* out) { out[0] = 0.0f; }

// ---- A-tile staging: 32 rows x 64 cols, threads 0..255 move 8 floats each ----
__device__ __forceinline__ void stageA_load(const float* __restrict__ L,
                                            int m0, int k0, bool full_chunk,
                                            int tid, v4f& v0, v4f& v1)
{
    v0 = (v4f){0.f, 0.f, 0.f, 0.f};
    v1 = (v4f){0.f, 0.f, 0.f, 0.f};
    const int row = tid >> 3;          // 0..31
    const int c8  = (tid & 7) << 3;    // 0,8,...,56
    const int gm  = m0 + row;
    if (gm < NV) {
        const size_t base = (size_t)gm * NV + k0 + c8;
        if (full_chunk) {
            // L is single-use: stream non-temporally, keep x resident in L2.
            v0 = __builtin_nontemporal_load((const v4f*)(L + base));
            v1 = __builtin_nontemporal_load((const v4f*)(L + base + 4));
        } else {
            #pragma unroll
            for (int j = 0; j < 4; ++j) {
                if (k0 + c8 + j < NV)     v0[j] = L[base + j];
                if (k0 + c8 + 4 + j < NV) v1[j] = L[base + 4 + j];
            }
        }
    }
}

__device__ __forceinline__ void stageA_store(float* __restrict__ buf, int tid,
                                             const v4f& v0, const v4f& v1)
{
    const int row = tid >> 3;
    const int c8  = (tid & 7) << 3;
    float* dst = &buf[row * ROWSTRIDE + c8];
    dst[0] = v0[0]; dst[1] = v0[1]; dst[2] = v0[2]; dst[3] = v0[3];
    dst[4] = v1[0]; dst[5] = v1[1]; dst[6] = v1[2]; dst[7] = v1[3];
}

// ---- 16 K-steps over a full chunk; each B fragment feeds 2 WMMAs ----
__device__ __forceinline__ void wmma_chunk_full(const float* __restrict__ buf,
                                                const float* __restrict__ xb,
                                                int k0, int m, int koff,
                                                v8f& acc0, v8f& acc1)
{
    #pragma unroll
    for (int kk = 0; kk < KB / 4; ++kk) {
        const float* ap0 = &buf[m * ROWSTRIDE + kk * 4 + koff];
        const float* ap1 = ap0 + 16 * ROWSTRIDE;
        v2f a0; a0.x = ap0[0]; a0.y = ap0[1];     // ds_load_b64
        v2f a1; a1.x = ap1[0]; a1.y = ap1[1];
        const int k = k0 + kk * 4 + koff;
        v2f b;
        b.x = xb[(size_t)(k + 0) * ND];
        b.y = xb[(size_t)(k + 1) * ND];
        acc0 = __builtin_amdgcn_wmma_f32_16x16x4_f32(
            false, a0, false, b, (short)0, acc0, false, false);
        acc1 = __builtin_amdgcn_wmma_f32_16x16x4_f32(
            false, a1, false, b, (short)0, acc1, false, false);
    }
}

// ---- Tail chunk: B indices clamped (A rows are zero there, product is 0) ----
__device__ __forceinline__ void wmma_chunk_tail(const float* __restrict__ buf,
                                                const float* __restrict__ xb,
                                                int k0, int m, int koff,
                                                v8f& acc0, v8f& acc1)
{
    #pragma unroll
    for (int kk = 0; kk < KB / 4; ++kk) {
        const float* ap0 = &buf[m * ROWSTRIDE + kk * 4 + koff];
        const float* ap1 = ap0 + 16 * ROWSTRIDE;
        v2f a0; a0.x = ap0[0]; a0.y = ap0[1];
        v2f a1; a1.x = ap1[0]; a1.y = ap1[1];
        const int k  = k0 + kk * 4 + koff;
        const int ka = (k + 0) < NV ? (k + 0) : (NV - 1);
        const int kb = (k + 1) < NV ? (k + 1) : (NV - 1);
        v2f b;
        b.x = xb[(size_t)ka * ND];
        b.y = xb[(size_t)kb * ND];
        acc0 = __builtin_amdgcn_wmma_f32_16x16x4_f32(
            false, a0, false, b, (short)0, acc0, false, false);
        acc1 = __builtin_amdgcn_wmma_f32_16x16x4_f32(
            false, a1, false, b, (short)0, acc1, false, false);
    }
}

__global__ __launch_bounds__(384)
void lap_loss_kernel(const float* __restrict__ x,   // [NB, NV, ND] f32
                     const float* __restrict__ L,   // [NV, NV] f32 row-major
                     float* __restrict__ out)       // scalar accumulator
{
    __shared__ float ldsA[2][MROWS * ROWSTRIDE];    // double-buffered A tiles
    __shared__ float sred[12];

    const int tid  = threadIdx.x;
    const int wave = tid >> 5;   // 0..11 -> N tile
    const int lane = tid & 31;
    const int m0   = blockIdx.x * MROWS;

    // Per-lane B addressing: B[k][n] with n = lane&15, k-offset (lane>>4)*2.
    // Column c = wave*16 + n maps to (b, d) of x[b][w][d].
    const int   n    = lane & 15;
    const int   koff = (lane >> 4) * 2;
    const int   col  = wave * 16 + n;          // 0..191
    const int   bidx = col / ND;
    const int   didx = col % ND;
    const float* xb  = x + (size_t)bidx * (NV * ND) + didx;

    // A addressing (16x4 f32 layout): lanes 0-15 hold M=lane, K = koff..koff+1
    const int m = lane & 15;

    v8f acc0 = {0.f, 0.f, 0.f, 0.f, 0.f, 0.f, 0.f, 0.f};
    v8f acc1 = {0.f, 0.f, 0.f, 0.f, 0.f, 0.f, 0.f, 0.f};

    // Prologue: stage chunk 0 into buffer 0.
    v4f s0, s1;
    if (tid < 256) {
        stageA_load(L, m0, 0, true, tid, s0, s1);
        stageA_store(ldsA[0], tid, s0, s1);
    }

    // Main loop: compute full chunks 0..NCHUNK-2, prefetch chunk i+1.
    // One barrier per iteration: it guarantees (a) buffer i&1 stores are
    // visible, and (b) all waves finished computing buffer (i+1)&1 last time,
    // so overwriting it below is safe.
    for (int i = 0; i < NCHUNK - 1; ++i) {
        __syncthreads();
        const int  knext     = (i + 1) * KB;
        const bool next_full = (i + 1) < (NCHUNK - 1);
        if (tid < 256) stageA_load(L, m0, knext, next_full, tid, s0, s1);
        wmma_chunk_full(ldsA[i & 1], xb, i * KB, m, koff, acc0, acc1);
        if (tid < 256) stageA_store(ldsA[(i + 1) & 1], tid, s0, s1);
    }

    // Tail chunk (42 valid K columns, zero-padded A, clamped B).
    __syncthreads();
    wmma_chunk_tail(ldsA[(NCHUNK - 1) & 1], xb, KTAIL, m, koff, acc0, acc1);

    // ---- Fused squared-norm reduction (zero-padded rows contribute 0) ----
    float s = 0.f;
    #pragma unroll
    for (int i = 0; i < 8; ++i) s += acc0[i] * acc0[i];
    #pragma unroll
    for (int i = 0; i < 8; ++i) s += acc1[i] * acc1[i];

    // wave32 butterfly reduce
    #pragma unroll
    for (int off = 16; off > 0; off >>= 1)
        s += __shfl_xor(s, off, 32);

    if (lane == 0) sred[wave] = s;
    __syncthreads();
    if (tid == 0) {
        float tot = 0.f;
        #pragma unroll
        for (int w = 0; w < 12; ++w) tot += sred[w];
        atomicAdd(out, tot * (1.0f / ((float)NB * (float)NV)));
    }
}

extern "C" void kernel_launch(void* const* d_in, const int* in_sizes, int n_in,
                              void* d_out, int out_size, void* d_ws, size_t ws_size,
                              hipStream_t stream) {
    const float* x = (const float*)d_in[0];   // [64, 6890, 3] f32
    const float* L = (const float*)d_in[1];   // [6890, 6890] f32
    float* out = (float*)d_out;               // scalar f32

    lap_zero_kernel<<<1, 1, 0, stream>>>(out);

    const int mtiles = (NV + MROWS - 1) / MROWS;  // 216
    lap_loss_kernel<<<mtiles, 384, 0, stream>>>(x, L, out);
}